// LinearLayer_77558519431745
// MI455X (gfx1250) — compile-verified
//
#include <hip/hip_runtime.h>

typedef __attribute__((ext_vector_type(2))) float v2f;
typedef __attribute__((ext_vector_type(4))) float v4f;
typedef __attribute__((ext_vector_type(8))) float v8f;
typedef __attribute__((ext_vector_type(4))) int   v4i;

#define FDIM 200  // features per example (reference: F = 200)

// One wave32 handles 16 examples. Reduction over the 200 features is done on
// the matrix pipe with V_WMMA_F32_16X16X4_F32 and an all-ones B matrix:
//   D[m][n] = sum_k A[m][k] * 1 + C   (exact fp32 row-sum)
// A-matrix (16x4 f32) VGPR layout per ISA 7.12.2:
//   lanes 0..15  : a[0]=A[m][0], a[1]=A[m][1]   (m = lane)
//   lanes 16..31 : a[0]=A[m][2], a[1]=A[m][3]   (m = lane-16)
// So lane (m, h=lane>>4) supplies products of features 8j+4h .. 8j+4h+3 per
// macro-iteration j via two WMMAs; union over h covers features 8j..8j+7 of
// example m exactly once. 200 = 25 * 8 -> no tail.
__global__ __launch_bounds__(256) void fm_linear_wmma(
    const int*   __restrict__ ids,   // [B, 200] int32
    const float* __restrict__ vals,  // [B, 200] f32
    const float* __restrict__ w,     // [1'000'000] f32 (L2-resident, 4 MB)
    const float* __restrict__ bias,  // [1] f32
    float*       __restrict__ out,   // [B] f32
    int nB)
{
  const int lane = threadIdx.x & 31;
  const int wave = threadIdx.x >> 5;
  const int m    = lane & 15;   // example within the wave's 16-row group
  const int h    = lane >> 4;   // which K-half of the A matrix this lane fills
  const int exBase = (blockIdx.x * 8 + wave) * 16;
  if (exBase >= nB) return;

  const long long rowOff = (long long)(exBase + m) * FDIM;
  const int*   idRow = ids  + rowOff;
  const float* vaRow = vals + rowOff;
  const int colBase = 4 * h;

  v8f acc = {};                       // C/D accumulator, 8 VGPRs
  const v2f ones = {1.0f, 1.0f};      // B = ones(4x16): every lane/VGPR = 1.0

  for (int j = 0; j < FDIM / 8; ++j) {
    const int c0 = 8 * j + colBase;
    // Coalesced 128-bit streams; 16B-aligned (row pitch 800 B, offsets %16==0)
    v4i id4 = *(const v4i*)(idRow + c0);
    v4f va4 = *(const v4f*)(vaRow + c0);
    // Irregular gathers: 4 MB table lives in L2 (192 MB) after first touch
    v2f a01 = { w[id4.x] * va4.x, w[id4.y] * va4.y };
    v2f a23 = { w[id4.z] * va4.z, w[id4.w] * va4.w };
    // 8 args: (neg_a, A, neg_b, B, c_mod, C, reuse_a, reuse_b)
    acc = __builtin_amdgcn_wmma_f32_16x16x4_f32(
        false, a01, false, ones, (short)0, acc, false, false);
    acc = __builtin_amdgcn_wmma_f32_16x16x4_f32(
        false, a23, false, ones, (short)0, acc, false, false);
  }

  // D layout: VGPR v, lanes 0..15 -> M=v (N=lane); lanes 16..31 -> M=8+v.
  // Every N column is identical, so lane 0 holds sums for examples 0..7 and
  // lane 16 holds sums for examples 8..15.
  if ((lane & 15) == 0) {
    const float b = bias[0];
    float* o = out + exBase + h * 8;   // 32B-aligned (exBase % 16 == 0)
    v4f lo = { acc[0] + b, acc[1] + b, acc[2] + b, acc[3] + b };
    v4f hi = { acc[4] + b, acc[5] + b, acc[6] + b, acc[7] + b };
    *(v4f*)(o)     = lo;
    *(v4f*)(o + 4) = hi;
  }
}

extern "C" void kernel_launch(void* const* d_in, const int* in_sizes, int n_in,
                              void* d_out, int out_size, void* d_ws, size_t ws_size,
                              hipStream_t stream) {
  const int*   ids  = (const int*)d_in[0];    // feat_ids  [B, 200] int32
  const float* vals = (const float*)d_in[1];  // feat_vals [B, 200] f32
  const float* w    = (const float*)d_in[2];  // weight_linear [1M, 1] f32
  const float* bias = (const float*)d_in[3];  // bias [1] f32
  float* out = (float*)d_out;                 // [B, 1] f32

  const int nB = in_sizes[0] / FDIM;          // 65536
  const int blocks = (nB + 127) / 128;        // 128 examples / block (8 waves)
  fm_linear_wmma<<<blocks, 256, 0, stream>>>(ids, vals, w, bias, out, nB);
}